// PreNet_58420145160556
// MI455X (gfx1250) — compile-verified
//
#include <hip/hip_runtime.h>
#include <hip/hip_bf16.h>

typedef __attribute__((ext_vector_type(2)))  float    v2f;
typedef __attribute__((ext_vector_type(8)))  float    v8f;
typedef __attribute__((ext_vector_type(4)))  int      v4i;

#if __has_builtin(__builtin_amdgcn_global_load_async_to_lds_b128) && \
    __has_builtin(__builtin_amdgcn_s_wait_asynccnt)
#define HAVE_ASYNC 1
typedef __attribute__((address_space(1))) v4i as1_v4i;   // global int4*
typedef __attribute__((address_space(3))) v4i as3_v4i;   // LDS int4*
#else
#define HAVE_ASYNC 0
#endif

// ---------------------------------------------------------------------------
// Stage 0: gather local grid points. local[n][gy][gx] = out[n,0,15(gy+1),15(gx+1)]
// ---------------------------------------------------------------------------
__global__ void k_gather(const float* __restrict__ out_in, float* __restrict__ local)
{
    const int n  = blockIdx.x;          // 0..1025
    const int t  = threadIdx.x;         // 0..255
    const int gy = t >> 4;
    const int gx = t & 15;
    local[(size_t)n * 256 + t] =
        out_in[(size_t)n * 65536 + (size_t)(15 * (gy + 1)) * 256 + (15 * (gx + 1))];
}

// ---------------------------------------------------------------------------
// Stage 1: c5a[k][o] = tanh(b74[o] + temp_row(k) . w74[o])   (temp row is
// local[k*256 .. k*256+768) because 3 consecutive frames are contiguous)
// ---------------------------------------------------------------------------
__global__ void k_fc74(const float* __restrict__ local, const float* __restrict__ w,
                       const float* __restrict__ b, float* __restrict__ c5a)
{
    __shared__ float row[768];
    const int k = blockIdx.x;           // 0..1023
    const int o = threadIdx.x;          // 0..63
    for (int i = o; i < 768; i += 64) row[i] = local[(size_t)k * 256 + i];
    __syncthreads();
    float acc = b[o];
    const float* wr = w + (size_t)o * 768;
    for (int i = 0; i < 768; ++i) acc += row[i] * wr[i];
    c5a[(size_t)k * 64 + o] = tanhf(acc);
}

// ---------------------------------------------------------------------------
// Stage 2: c5b[k][o] = tanh(b75[o] + c5a[k] . w75[o])
// ---------------------------------------------------------------------------
__global__ void k_fc75(const float* __restrict__ c5a, const float* __restrict__ w,
                       const float* __restrict__ b, float* __restrict__ c5b)
{
    __shared__ float row[64];
    const int k = blockIdx.x;
    const int o = threadIdx.x;
    row[o] = c5a[(size_t)k * 64 + o];
    __syncthreads();
    float acc = b[o];
    const float* wr = w + (size_t)o * 64;
#pragma unroll 8
    for (int i = 0; i < 64; ++i) acc += row[i] * wr[i];
    c5b[(size_t)k * 64 + o] = tanhf(acc);
}

// ---------------------------------------------------------------------------
// Stage 3 (dominant): ref2[j] = 340 + 1160/1024 * sum_k sigmoid(c5b[k].w76[j]+b76[j])
//
// Block = 256 threads = 8 waves; each wave owns a 16-column tile (block: 128
// cols, grid: 512 blocks). B tile (16 cols x 64 K) lives in 32 VGPRs per wave.
// A (= c5b) is staged through double-buffered LDS in 64-row chunks shared by
// all 8 waves. Next chunk is fetched with GLOBAL_LOAD_ASYNC_TO_LDS_B128
// (ASYNCcnt-tracked, zero VGPR staging) issued before the WMMA work, so the
// copy overlaps 64 WMMAs and there is no register-prefetch spill pressure.
// LDS row stride padded to 68 floats -> conflict-free ds_load_b64 A reads.
// ---------------------------------------------------------------------------
#define LSTRIDE 68

__global__ void __launch_bounds__(256) k_fc76_reduce(
    const float* __restrict__ c5b, const float* __restrict__ w76,
    const float* __restrict__ b76, float* __restrict__ ref2)
{
    __shared__ float Abuf[2][64][LSTRIDE];          // 2 x 64 x 68 x 4B = 34 KB

    const int tid  = threadIdx.x;
    const int lane = tid & 31;
    const int wv   = tid >> 5;                      // wave in block: 0..7
    const int col0 = blockIdx.x * 128 + wv * 16;    // 16 cols per wave
    const int half = lane >> 4;                     // 0 or 1
    const int nn   = lane & 15;                     // col within tile / M row
    const int kbase = half * 2;                     // f32 WMMA K split 0/1 vs 2/3

    // ---- B tile: 16 slices of 4x16, ISA layout lane->(K=2*half+v, N=lane%16)
    const float  bias = b76[col0 + nn];
    const float* wrow = w76 + (size_t)(col0 + nn) * 64;
    v2f B[16];
#pragma unroll
    for (int kk = 0; kk < 16; ++kk) {
        B[kk][0] = wrow[4 * kk + kbase];
        B[kk][1] = wrow[4 * kk + kbase + 1];
    }

    float ssum = 0.0f;

    // ---- prologue: stage chunk 0 (rows 0..63) into buffer 0
#if HAVE_ASYNC
#pragma unroll
    for (int i = 0; i < 4; ++i) {
        const int idx = i * 256 + tid;              // float4 index 0..1023
        const int row = idx >> 4;
        const int col = (idx & 15) << 2;
        __builtin_amdgcn_global_load_async_to_lds_b128(
            (as1_v4i*)const_cast<float*>(c5b + (size_t)idx * 4),
            (as3_v4i*)&Abuf[0][row][col], 0, 0);
    }
    __builtin_amdgcn_s_wait_asynccnt(0);
#else
    {
        const float4* src = (const float4*)c5b;
#pragma unroll
        for (int i = 0; i < 4; ++i) {
            const int idx = i * 256 + tid;
            const float4 r = src[idx];
            *(float4*)&Abuf[0][idx >> 4][(idx & 15) << 2] = r;
        }
    }
#endif
    __syncthreads();

    int buf = 0;
    for (int c = 0; c < 16; ++c) {                  // 16 chunks x 64 rows = 1024
#if HAVE_ASYNC
        // ---- async-copy next chunk straight into the other LDS buffer;
        //      overlaps the 64 WMMAs below, no VGPR staging at all.
        if (c < 15) {
            const float* gsrc = c5b + (size_t)(c + 1) * 64 * 64;
#pragma unroll
            for (int i = 0; i < 4; ++i) {
                const int idx = i * 256 + tid;
                const int row = idx >> 4;
                const int col = (idx & 15) << 2;
                __builtin_amdgcn_global_load_async_to_lds_b128(
                    (as1_v4i*)const_cast<float*>(gsrc + (size_t)idx * 4),
                    (as3_v4i*)&Abuf[buf ^ 1][row][col], 0, 0);
            }
        }
#endif

        // ---- compute: 4 row-tiles of 16 from LDS chunk `buf` (2-way interleave)
        const float* Ab = &Abuf[buf][0][0];
#pragma unroll 2
        for (int t = 0; t < 4; ++t) {
            const float* ar = Ab + (t * 16 + nn) * LSTRIDE;
            v2f A[16];
#pragma unroll
            for (int kk = 0; kk < 16; ++kk) {       // ds_load_2addr_b64 pairs
                A[kk][0] = ar[4 * kk + kbase];
                A[kk][1] = ar[4 * kk + kbase + 1];
            }
            v8f acc = {};
#pragma unroll
            for (int kk = 0; kk < 16; ++kk)         // WMMA chain (K = 64)
                acc = __builtin_amdgcn_wmma_f32_16x16x4_f32(
                    false, A[kk], false, B[kk], (short)0, acc, false, false);
#pragma unroll
            for (int r = 0; r < 8; ++r) {           // fused sigmoid + batch sum
                const float x = acc[r] + bias;      // row M = r + 8*half
                ssum += 1.0f / (1.0f + __expf(-x));
            }
        }

#if HAVE_ASYNC
        __builtin_amdgcn_s_wait_asynccnt(0);        // this wave's copies landed
#else
        // ---- fallback: copy next chunk after compute (regs live only briefly)
        if (c < 15) {
            const float4* src = (const float4*)(c5b + (size_t)(c + 1) * 64 * 64);
#pragma unroll
            for (int i = 0; i < 4; ++i) {
                const int idx = i * 256 + tid;
                const float4 r = src[idx];
                *(float4*)&Abuf[buf ^ 1][idx >> 4][(idx & 15) << 2] = r;
            }
        }
#endif
        __syncthreads();                            // all waves' data visible
        buf ^= 1;
    }

    // lanes l and l+16 hold the same column (rows r and r+8): merge halves
    ssum += __shfl_xor(ssum, 16, 32);
    if (half == 0)
        ref2[col0 + nn] = 340.0f + 1160.0f * ssum * (1.0f / 1024.0f);
}

// ---------------------------------------------------------------------------
// Stage 4: 5x5 conv with circular padding (+ bias) on the 256x256 mean map.
// ---------------------------------------------------------------------------
__global__ void k_conv(const float* __restrict__ ref2, const float* __restrict__ cw,
                       const float* __restrict__ cb, float* __restrict__ out)
{
    const int idx = blockIdx.x * blockDim.x + threadIdx.x;   // 0..65535
    const int y = idx >> 8;
    const int x = idx & 255;
    float acc = cb[0];
#pragma unroll
    for (int dy = 0; dy < 5; ++dy) {
        const int yy = (y + dy + 254) & 255;                 // (y + dy - 2) mod 256
#pragma unroll
        for (int dx = 0; dx < 5; ++dx) {
            const int xx = (x + dx + 254) & 255;
            acc += cw[dy * 5 + dx] * ref2[yy * 256 + xx];
        }
    }
    out[idx] = acc;
}

// ---------------------------------------------------------------------------
// Host-side launch
// ---------------------------------------------------------------------------
extern "C" void kernel_launch(void* const* d_in, const int* in_sizes, int n_in,
                              void* d_out, int out_size, void* d_ws, size_t ws_size,
                              hipStream_t stream)
{
    (void)in_sizes; (void)n_in; (void)out_size; (void)ws_size;

    const float* out_in = (const float*)d_in[0];   // [1026,1,256,256]
    const float* w74    = (const float*)d_in[1];   // [64,768]
    const float* b74    = (const float*)d_in[2];   // [64]
    const float* w75    = (const float*)d_in[3];   // [64,64]
    const float* b75    = (const float*)d_in[4];   // [64]
    const float* w76    = (const float*)d_in[5];   // [65536,64]
    const float* b76    = (const float*)d_in[6];   // [65536]
    const float* cw     = (const float*)d_in[7];   // [1,1,5,5]
    const float* cb     = (const float*)d_in[8];   // [1]
    float* out = (float*)d_out;                    // [1,1,256,256]

    char*  ws    = (char*)d_ws;
    float* local = (float*)(ws);                                  // 1026*256 f32
    float* c5a   = (float*)(ws + 1050624);                        // 1024*64  f32
    float* c5b   = (float*)(ws + 1050624 + 262144);               // 1024*64  f32
    float* ref2  = (float*)(ws + 1050624 + 2 * 262144);           // 65536    f32

    k_gather<<<1026, 256, 0, stream>>>(out_in, local);
    k_fc74<<<1024, 64, 0, stream>>>(local, w74, b74, c5a);
    k_fc75<<<1024, 64, 0, stream>>>(c5a, w75, b75, c5b);
    k_fc76_reduce<<<512, 256, 0, stream>>>(c5b, w76, b76, ref2);
    k_conv<<<256, 256, 0, stream>>>(ref2, cw, cb, out);
}